// DSD_49357764166126
// MI455X (gfx1250) — compile-verified
//
#include <hip/hip_runtime.h>
#include <hip/hip_bf16.h>
#include <math.h>

typedef __attribute__((ext_vector_type(2))) float v2f;
typedef __attribute__((ext_vector_type(8))) float v8f;

#define BATCH    32
#define SEQ_LEN  720
#define PRED_LEN 720
#define CH       862
#define NFREQ    361          // rfft bins (720/2+1)
#define NMAG     360          // bins excluding DC
#define FP2      736          // 2*NFREQ=722 padded to x16
#define ROWS     (BATCH*CH)   // 27584
#define HID_FREQ 64
#define HID_ALL  128
#define LC_HID   32
#define LIN_LEN  361

// GEMM epilogue flags
#define GF_ACC   1   // C += result (read-modify-write)
#define GF_RELU  2   // relu after bias/acc
#define GF_TOUT  4   // transposed store into (b, n, c) output slab

// ---------------------------------------------------------------------------
// CDNA5 async global->LDS copies (ASYNCcnt path). The builtin signature is
// void(v4i AS(1)* src, v4i AS(3)* dst, imm int offset, imm int cpol) — param
// types confirmed by the compiler diagnostic (b128 payload = int vec16 bytes).
// ---------------------------------------------------------------------------
#if defined(__AMDGCN__) && \
    __has_builtin(__builtin_amdgcn_global_load_async_to_lds_b128) && \
    __has_builtin(__builtin_amdgcn_s_wait_asynccnt)
#define HAVE_ASYNC_LDS 1
typedef int v4i_vs __attribute__((vector_size(16)));
typedef __attribute__((address_space(1))) v4i_vs g_v4i;
typedef __attribute__((address_space(3))) v4i_vs l_v4i;
__device__ __forceinline__ void async_copy_b128(const void* g, void* l)
{
  // flat LDS pointer: addr[31:0] is the LDS byte offset (aperture in high bits)
  __builtin_amdgcn_global_load_async_to_lds_b128(
      (g_v4i*)(uintptr_t)g, (l_v4i*)(uint32_t)(uintptr_t)l, 0, 0);
}
#else
#define HAVE_ASYNC_LDS 0
#endif

// ---------------------------------------------------------------------------
// f32 WMMA GEMM: C(MxN) = A(MxK) @ B(KxN) [+bias[n]] [+C] [relu]
// 128 threads = 4 waves; workgroup tile 64(M) x (16*NSUB)(N); K step 16.
// Each wave owns 16 rows x NSUB 16-col tiles -> 4*NSUB wmma per K step.
// Double-buffered LDS stages fed by async global->LDS when available.
// Requires M%64==0, K%16==0; N handled as full-64 blocks + one remainder.
// ---------------------------------------------------------------------------
template<int NSUB>
__global__ __launch_bounds__(128) void gemm_f32_wmma(
    const float* __restrict__ A, const float* __restrict__ B,
    float* __restrict__ C, const float* __restrict__ bias,
    int M, int N, int K, int lda, int ldb, int ldc, int flags, int n_base)
{
  // strides: 20 (=80B) and 68 (=272B): rows 16B-aligned for async B128 stores,
  // and 16-lane fragment reads remain bank-conflict-free (gcd-safe offsets).
  __shared__ float As[2][64][20];
  __shared__ float Bs[2][16][68];

  const int t    = threadIdx.x;
  const int lane = t & 31;
  const int wv   = t >> 5;            // wave id 0..3 -> M sub-tile
  const int half = lane >> 4;         // 0: lanes 0-15, 1: lanes 16-31
  const int l16  = lane & 15;
  const int m0   = blockIdx.x * 64;
  const int n0   = n_base + blockIdx.y * 64;

  // stage loader: A tile 64x16, B tile 16x(16*NSUB); 4 x B128 per thread
  auto load_stage = [&](int bi, int k0) {
    const int ra = t >> 2, qa = (t & 3) * 4;         // A slots t, t+128
    const int rb = t >> 4, cb = (t & 15) * 4;        // B slots t, t+128
#if HAVE_ASYNC_LDS
    async_copy_b128(A + (size_t)(m0 + ra)      * lda + k0 + qa, &As[bi][ra]     [qa]);
    async_copy_b128(A + (size_t)(m0 + ra + 32) * lda + k0 + qa, &As[bi][ra + 32][qa]);
    if (cb < NSUB * 16) {
      async_copy_b128(B + (size_t)(k0 + rb)     * ldb + n0 + cb, &Bs[bi][rb]    [cb]);
      async_copy_b128(B + (size_t)(k0 + rb + 8) * ldb + n0 + cb, &Bs[bi][rb + 8][cb]);
    }
#else
    float4 a0 = *(const float4*)(A + (size_t)(m0 + ra)      * lda + k0 + qa);
    float4 a1 = *(const float4*)(A + (size_t)(m0 + ra + 32) * lda + k0 + qa);
    As[bi][ra][qa+0]=a0.x; As[bi][ra][qa+1]=a0.y; As[bi][ra][qa+2]=a0.z; As[bi][ra][qa+3]=a0.w;
    As[bi][ra+32][qa+0]=a1.x; As[bi][ra+32][qa+1]=a1.y; As[bi][ra+32][qa+2]=a1.z; As[bi][ra+32][qa+3]=a1.w;
    if (cb < NSUB * 16) {
      float4 b0 = *(const float4*)(B + (size_t)(k0 + rb)     * ldb + n0 + cb);
      float4 b1 = *(const float4*)(B + (size_t)(k0 + rb + 8) * ldb + n0 + cb);
      Bs[bi][rb][cb+0]=b0.x; Bs[bi][rb][cb+1]=b0.y; Bs[bi][rb][cb+2]=b0.z; Bs[bi][rb][cb+3]=b0.w;
      Bs[bi][rb+8][cb+0]=b1.x; Bs[bi][rb+8][cb+1]=b1.y; Bs[bi][rb+8][cb+2]=b1.z; Bs[bi][rb+8][cb+3]=b1.w;
    }
#endif
  };

  v8f acc[NSUB];
  #pragma unroll
  for (int ns = 0; ns < NSUB; ++ns)
    acc[ns] = (v8f){0.f,0.f,0.f,0.f,0.f,0.f,0.f,0.f};

  const int nk = K / 16;
  load_stage(0, 0);

  for (int kk = 0; kk < nk; ++kk) {
    const int cur = kk & 1;
    if (kk + 1 < nk) {
      load_stage(cur ^ 1, (kk + 1) * 16);
#if HAVE_ASYNC_LDS
      __builtin_amdgcn_s_wait_asynccnt(4);   // leave next stage in flight
#endif
    } else {
#if HAVE_ASYNC_LDS
      __builtin_amdgcn_s_wait_asynccnt(0);
#endif
    }
    __syncthreads();

    // 16x(16*NSUB)x16 via 4*NSUB x V_WMMA_F32_16X16X4_F32; A frag reused
    #pragma unroll
    for (int s4 = 0; s4 < 16; s4 += 4) {
      v2f av;
      // A 16x4 layout: VGPR0 = K {s4, s4+2}(half), VGPR1 = K {s4+1, s4+3}
      av.x = As[cur][wv*16 + l16][s4 + 2*half + 0];
      av.y = As[cur][wv*16 + l16][s4 + 2*half + 1];
      #pragma unroll
      for (int ns = 0; ns < NSUB; ++ns) {
        v2f bv;
        bv.x = Bs[cur][s4 + 2*half + 0][ns*16 + l16];
        bv.y = Bs[cur][s4 + 2*half + 1][ns*16 + l16];
        acc[ns] = __builtin_amdgcn_wmma_f32_16x16x4_f32(
            false, av, false, bv, (short)0, acc[ns], false, false);
      }
    }
    __syncthreads();
  }

  // Epilogue. C/D layout: VGPR r holds (M=r, N=l16) lanes 0-15, (M=r+8) 16-31.
  const int mb = m0 + wv*16 + half*8;
  #pragma unroll
  for (int ns = 0; ns < NSUB; ++ns) {
    const int n = n0 + ns*16 + l16;
    const float bval = bias ? bias[n] : 0.f;
    if (flags & GF_TOUT) {
      // scatter into output slab laid out (b, n, c), row m = b*CH + c
      #pragma unroll
      for (int r = 0; r < 8; ++r) {
        int m  = mb + r;
        int bb = m / CH;
        int cc = m - bb * CH;
        C[(size_t)bb * N * CH + (size_t)n * CH + cc] = acc[ns][r] + bval;
      }
    } else {
      #pragma unroll
      for (int r = 0; r < 8; ++r) {
        size_t off = (size_t)(mb + r) * ldc + n;
        float v = acc[ns][r] + bval;
        if (flags & GF_ACC)  v += C[off];
        if (flags & GF_RELU) v = v > 0.f ? v : 0.f;
        C[off] = v;
      }
    }
  }
}

// host-side dispatcher: full 64-wide column blocks + one remainder block
static void launch_gemm(const float* A, const float* B, float* C, const float* bias,
                        int M, int N, int K, int lda, int ldb, int ldc, int flags,
                        hipStream_t stream)
{
  const int nfull = N / 64;
  const int rem   = N - nfull * 64;
  if (nfull)
    gemm_f32_wmma<4><<<dim3(M/64, nfull), 128, 0, stream>>>(
        A, B, C, bias, M, N, K, lda, ldb, ldc, flags, 0);
  if (rem == 32)
    gemm_f32_wmma<2><<<dim3(M/64, 1), 128, 0, stream>>>(
        A, B, C, bias, M, N, K, lda, ldb, ldc, flags, nfull*64);
  else if (rem == 16)
    gemm_f32_wmma<1><<<dim3(M/64, 1), 128, 0, stream>>>(
        A, B, C, bias, M, N, K, lda, ldb, ldc, flags, nfull*64);
  else if (rem == 48)
    gemm_f32_wmma<3><<<dim3(M/64, 1), 128, 0, stream>>>(
        A, B, C, bias, M, N, K, lda, ldb, ldc, flags, nfull*64);
}

// ---------------------------------------------------------------------------
// Transpose batch_x (B, L, C) -> XT (B*C, L) with LDS tiles
// ---------------------------------------------------------------------------
__global__ void transpose_x(const float* __restrict__ x, float* __restrict__ xt)
{
  __shared__ float tile[32][33];
  const int b  = blockIdx.z;
  const int l0 = blockIdx.x * 32, c0 = blockIdx.y * 32;
  const int tx = threadIdx.x, ty = threadIdx.y;
  #pragma unroll
  for (int j = 0; j < 32; j += 8) {
    int l = l0 + ty + j, c = c0 + tx;
    if (l < SEQ_LEN && c < CH)
      tile[ty + j][tx] = x[(size_t)b*SEQ_LEN*CH + (size_t)l*CH + c];
  }
  __syncthreads();
  #pragma unroll
  for (int j = 0; j < 32; j += 8) {
    int c = c0 + ty + j, l = l0 + tx;
    if (l < SEQ_LEN && c < CH)
      xt[((size_t)b*CH + c)*SEQ_LEN + l] = tile[tx][ty + j];
  }
}

// ---------------------------------------------------------------------------
// DFT matrices (rfft / irfft as GEMM), generated on device.
// D (720 x 736): cols 2f = cos, 2f+1 = -sin ; padded cols zero.
// DI (736 x 720): rows 2f = w*cos/720, 2f+1 = -w*sin/720 ; padded rows zero.
// ---------------------------------------------------------------------------
__global__ void gen_dft(float* __restrict__ D)
{
  int idx = blockIdx.x * blockDim.x + threadIdx.x;
  if (idx >= SEQ_LEN * FP2) return;
  int l = idx / FP2, col = idx - l * FP2;
  float v = 0.f;
  if (col < 2*NFREQ) {
    int f = col >> 1;
    int m = (l * f) % SEQ_LEN;
    float ang = 6.283185307179586f * (float)m / (float)SEQ_LEN;
    v = (col & 1) ? -sinf(ang) : cosf(ang);
  }
  D[idx] = v;
}

__global__ void gen_idft(float* __restrict__ DI)
{
  int idx = blockIdx.x * blockDim.x + threadIdx.x;
  if (idx >= FP2 * SEQ_LEN) return;
  int r = idx / SEQ_LEN, l = idx - r * SEQ_LEN;
  float v = 0.f;
  if (r < 2*NFREQ) {
    int f = r >> 1;
    float w = (f == 0 || f == NMAG) ? 1.f : 2.f;
    int m = (l * f) % SEQ_LEN;
    float ang = 6.283185307179586f * (float)m / (float)SEQ_LEN;
    float s = w / (float)SEQ_LEN;
    v = (r & 1) ? -s * sinf(ang) : s * cosf(ang);
  }
  DI[idx] = v;
}

// dst[k*N+n] = src[n*sld + soff + k]  (weight transpose for GEMM B operand)
__global__ void transpose_w(float* __restrict__ dst, const float* __restrict__ src,
                            int N, int K, int sld, int soff)
{
  int idx = blockIdx.x * blockDim.x + threadIdx.x;
  if (idx >= N * K) return;
  int k = idx / N, n = idx - k * N;
  dst[idx] = src[(size_t)n * sld + soff + k];
}

// ---------------------------------------------------------------------------
// Per-(b,c) frequency filter: magnitude sort (bitonic), Lorenz/Gini, tiny MLP,
// sigmoid energy threshold, unsort, apply. One 512-thread WG per spectrum row.
// XF row (FP2 floats, re/im interleaved, f=0..360) updated in place;
// XFILT row receives xf*filt.
// ---------------------------------------------------------------------------
__global__ __launch_bounds__(512) void filter_kernel(
    float* __restrict__ XF, float* __restrict__ XFILT,
    const float* __restrict__ w1, const float* __restrict__ b1,
    const float* __restrict__ w2, const float* __restrict__ b2)
{
  __shared__ float s_key[512];
  __shared__ int   s_idx[512];
  __shared__ float s_cum[512];
  __shared__ float s_red[512];
  __shared__ float s_lor[LIN_LEN];   // lorenz_pad, [0] = 0
  __shared__ float s_filt[NFREQ];
  __shared__ float s_h[LC_HID];
  __shared__ float s_thr[1];

  const int t = threadIdx.x;
  float* xfrow = XF    + (size_t)blockIdx.x * FP2;
  float* xorow = XFILT + (size_t)blockIdx.x * FP2;

  // 1) magnitudes of bins 1..360 (pad with +inf so pads sort to the top)
  if (t < NMAG) {
    float re = xfrow[2*(t+1)], im = xfrow[2*(t+1)+1];
    s_key[t] = sqrtf(re*re + im*im);
    s_idx[t] = t;
  } else {
    s_key[t] = 3.4028235e38f;
    s_idx[t] = t;
  }
  __syncthreads();

  // 2) bitonic ascending key-value sort over 512
  for (int k = 2; k <= 512; k <<= 1) {
    for (int j = k >> 1; j > 0; j >>= 1) {
      int ixj = t ^ j;
      if (ixj > t) {
        bool up = ((t & k) == 0);
        float ka = s_key[t], kb = s_key[ixj];
        bool sw = up ? (ka > kb) : (ka < kb);
        if (sw) {
          s_key[t] = kb; s_key[ixj] = ka;
          int ia = s_idx[t]; s_idx[t] = s_idx[ixj]; s_idx[ixj] = ia;
        }
      }
      __syncthreads();
    }
  }

  // 3) inclusive scan of sorted magnitudes (ascending cumsum)
  s_cum[t] = (t < NMAG) ? s_key[t] : 0.f;
  __syncthreads();
  for (int off = 1; off < 512; off <<= 1) {
    float add = (t >= off) ? s_cum[t - off] : 0.f;
    __syncthreads();
    s_cum[t] += add;
    __syncthreads();
  }
  const float total = s_cum[NMAG-1];
  const float tot   = fmaxf(total, 1e-8f);

  // 4) Lorenz curve + sum for Gini
  if (t == 0) s_lor[0] = 0.f;
  if (t < NMAG) s_lor[t+1] = s_cum[t] / tot;
  s_red[t] = (t < NMAG) ? s_cum[t] : 0.f;
  __syncthreads();
  for (int off = 256; off > 0; off >>= 1) {
    if (t < off) s_red[t] += s_red[t + off];
    __syncthreads();
  }
  const float gini = 1.f - 2.f * (s_red[0] / tot) / (float)NMAG;

  // 5) Lorenz MLP: h = relu(W1 @ lorenz_pad + b1), g = tanh(w2 . h + b2)
  if (t < LC_HID) {
    float a = b1[t];
    const float* wr = w1 + (size_t)t * LIN_LEN;
    for (int f = 1; f < LIN_LEN; ++f) a += wr[f] * s_lor[f];
    s_h[t] = fmaxf(a, 0.f);
  }
  __syncthreads();
  if (t == 0) {
    float g = b2[0];
    #pragma unroll
    for (int j = 0; j < LC_HID; ++j) g += w2[j] * s_h[j];
    g = tanhf(g);
    float et = gini + g;
    et = fminf(fmaxf(et, 0.f), 1.f);
    s_thr[0] = et * total;
  }
  __syncthreads();

  // 6) descending-order sigmoid filter, scatter back to original bins
  if (t < NMAG) {
    // desc rank t -> asc position 359-t
    int ap = NMAG - 1 - t;
    float cum_desc = total - s_cum[ap] + s_key[ap];
    float sf = 1.f / (1.f + expf(-(s_thr[0] - cum_desc)));
    s_filt[s_idx[ap] + 1] = sf;
  }
  if (t == 0) s_filt[0] = 1.f;  // DC passes
  __syncthreads();

  // 7) apply: XFILT = xf*filt ; xf *= (1-filt) ; zero XFILT pad columns
  if (t < NFREQ) {
    float re = xfrow[2*t], im = xfrow[2*t+1];
    float fl = s_filt[t];
    xorow[2*t]   = re * fl;
    xorow[2*t+1] = im * fl;
    xfrow[2*t]   = re * (1.f - fl);
    xfrow[2*t+1] = im * (1.f - fl);
  } else if (t < NFREQ + (FP2 - 2*NFREQ)/2) {
    int c = 2*NFREQ + 2*(t - NFREQ);
    xorow[c] = 0.f; xorow[c+1] = 0.f;
  }
}

// ---------------------------------------------------------------------------
extern "C" void kernel_launch(void* const* d_in, const int* in_sizes, int n_in,
                              void* d_out, int out_size, void* d_ws, size_t ws_size,
                              hipStream_t stream)
{
  const float* x      = (const float*)d_in[0];
  const float* lc_w1  = (const float*)d_in[1];
  const float* lc_b1  = (const float*)d_in[2];
  const float* lc_w2  = (const float*)d_in[3];
  const float* lc_b2  = (const float*)d_in[4];
  const float* mf_wf  = (const float*)d_in[5];
  const float* mf_bf  = (const float*)d_in[6];
  const float* mf_wa1 = (const float*)d_in[7];
  const float* mf_ba1 = (const float*)d_in[8];
  const float* mf_wa2 = (const float*)d_in[9];
  const float* mf_ba2 = (const float*)d_in[10];
  float* out = (float*)d_out;
  const size_t slab = (size_t)BATCH * PRED_LEN * CH;

  // workspace carve (f32)
  float* p     = (float*)d_ws;
  float* XT    = p; p += (size_t)ROWS * SEQ_LEN;    // x transposed (B*C, L)
  float* XF    = p; p += (size_t)ROWS * FP2;        // spectrum (re/im interleaved)
  float* XFILT = p; p += (size_t)ROWS * FP2;        // filtered spectrum
  float* FILT  = p; p += (size_t)ROWS * SEQ_LEN;    // irfft(filtered) (B*C, L)
  float* HF    = p; p += (size_t)ROWS * HID_FREQ;
  float* H2    = p; p += (size_t)ROWS * HID_ALL;
  float* Dm    = p; p += (size_t)SEQ_LEN * FP2;     // DFT matrix
  float* DIm   = p; p += (size_t)FP2 * SEQ_LEN;     // iDFT matrix
  float* WFT   = p; p += (size_t)SEQ_LEN * HID_FREQ;
  float* WA1AT = p; p += (size_t)HID_FREQ * HID_ALL;
  float* WA1BT = p; p += (size_t)SEQ_LEN * HID_ALL;
  float* WA2T  = p; p += (size_t)HID_ALL * PRED_LEN;

  // 1) transpose input, build DFT/iDFT matrices
  {
    dim3 g((SEQ_LEN+31)/32, (CH+31)/32, BATCH), b(32,8);
    transpose_x<<<g, b, 0, stream>>>(x, XT);
  }
  gen_dft <<<(SEQ_LEN*FP2 + 255)/256, 256, 0, stream>>>(Dm);
  gen_idft<<<(FP2*SEQ_LEN + 255)/256, 256, 0, stream>>>(DIm);

  // 2) rfft as GEMM: XF = XT @ D
  launch_gemm(XT, Dm, XF, nullptr, ROWS, FP2, SEQ_LEN, SEQ_LEN, FP2, FP2, 0, stream);

  for (int i = 0; i < 2; ++i) {
    // 3) sort/Lorenz/MLP/threshold filter; updates XF, writes XFILT
    filter_kernel<<<ROWS, 512, 0, stream>>>(
        XF, XFILT,
        lc_w1 + (size_t)i*LC_HID*LIN_LEN, lc_b1 + (size_t)i*LC_HID,
        lc_w2 + (size_t)i*LC_HID,         lc_b2 + i);

    // 4) irfft of filtered band: FILT = XFILT @ DI
    launch_gemm(XFILT, DIm, FILT, nullptr, ROWS, SEQ_LEN, FP2, FP2, SEQ_LEN,
                SEQ_LEN, 0, stream);

    // 5) hf = relu(FILT @ wf^T + bf)
    transpose_w<<<(HID_FREQ*SEQ_LEN + 255)/256, 256, 0, stream>>>(
        WFT, mf_wf + (size_t)i*HID_FREQ*SEQ_LEN, HID_FREQ, SEQ_LEN, SEQ_LEN, 0);
    launch_gemm(FILT, WFT, HF, mf_bf + (size_t)i*HID_FREQ,
                ROWS, HID_FREQ, SEQ_LEN, SEQ_LEN, HID_FREQ, HID_FREQ,
                GF_RELU, stream);

    // 6) h2 = relu([hf | x] @ wa1^T + ba1) as two accumulating GEMMs
    transpose_w<<<(HID_ALL*HID_FREQ + 255)/256, 256, 0, stream>>>(
        WA1AT, mf_wa1 + (size_t)i*HID_ALL*(HID_FREQ+SEQ_LEN),
        HID_ALL, HID_FREQ, HID_FREQ+SEQ_LEN, 0);
    transpose_w<<<(HID_ALL*SEQ_LEN + 255)/256, 256, 0, stream>>>(
        WA1BT, mf_wa1 + (size_t)i*HID_ALL*(HID_FREQ+SEQ_LEN),
        HID_ALL, SEQ_LEN, HID_FREQ+SEQ_LEN, HID_FREQ);
    launch_gemm(HF, WA1AT, H2, nullptr, ROWS, HID_ALL, HID_FREQ,
                HID_FREQ, HID_ALL, HID_ALL, 0, stream);
    launch_gemm(XT, WA1BT, H2, mf_ba1 + (size_t)i*HID_ALL,
                ROWS, HID_ALL, SEQ_LEN, SEQ_LEN, HID_ALL, HID_ALL,
                GF_ACC | GF_RELU, stream);

    // 7) pred = h2 @ wa2^T + ba2, stored transposed straight into out[1+i]
    transpose_w<<<(PRED_LEN*HID_ALL + 255)/256, 256, 0, stream>>>(
        WA2T, mf_wa2 + (size_t)i*PRED_LEN*HID_ALL, PRED_LEN, HID_ALL, HID_ALL, 0);
    launch_gemm(H2, WA2T, out + (size_t)(1+i)*slab, mf_ba2 + (size_t)i*PRED_LEN,
                ROWS, PRED_LEN, HID_ALL, HID_ALL, PRED_LEN, 0, GF_TOUT, stream);
  }

  // 8) x_res = irfft(residual spectrum), transposed straight into out[0]
  launch_gemm(XF, DIm, out, nullptr, ROWS, SEQ_LEN, FP2, FP2, SEQ_LEN, 0,
              GF_TOUT, stream);
}